// GenModel_32744830665487
// MI455X (gfx1250) — compile-verified
//
#include <hip/hip_runtime.h>
#include <stdint.h>
#include <math.h>

#define NSAMP 4096
#define DDIM  3
#define HF    24
#define HG    32
#define GRP   8             // lanes cooperating on one sample (divides 24 and 32)
#define TPB   256           // 8 waves
#define SPB   (TPB / GRP)   // 32 samples per block
#define TCHUNK 128          // steps of noise per TDM tile
#define ROWW (SPB * DDIM)   // 96 floats of noise per step per block

typedef unsigned int u32;
typedef u32 u32x4 __attribute__((ext_vector_type(4)));
typedef int i32x4 __attribute__((ext_vector_type(4)));
typedef int i32x8 __attribute__((ext_vector_type(8)));

// Issue a 2-D TDM tile load: tile_d1 rows of tile_d0 f32 elements, row stride
// stride_d0 elements, from gaddr into LDS offset lds_off. Wave-level op,
// tracked with TENSORcnt.
__device__ inline void tdm_load_2d(u32 lds_off, const float* gaddr,
                                   u32 tile_d0, u32 tile_d1,
                                   u32 tensor_d0, u32 tensor_d1,
                                   u32 stride_d0) {
  unsigned long long ga = (unsigned long long)(uintptr_t)gaddr;
  u32x4 g0;
  g0.x = 1u;                                   // count=1, user descriptor
  g0.y = lds_off;                              // lds_addr [63:32]
  g0.z = (u32)(ga & 0xFFFFFFFFull);            // global_addr[31:0]
  g0.w = (u32)((ga >> 32) & 0x1FFFFFFull)      // global_addr[56:32]
         | (2u << 30);                         // type=2 ("image")
  i32x8 g1;
  g1[0] = (int)(2u << 16);                     // data_size=2 (4B), wg_mask=0
  g1[1] = (int)((tensor_d0 & 0xFFFFu) << 16);  // tensor_dim0[15:0]
  g1[2] = (int)((tensor_d0 >> 16) | ((tensor_d1 & 0xFFFFu) << 16));
  g1[3] = (int)((tensor_d1 >> 16) | (tile_d0 << 16));   // tile_dim0 @ [127:112]
  g1[4] = (int)(tile_d1 & 0xFFFFu);            // tile_dim1; tile_dim2=0
  g1[5] = (int)stride_d0;                      // tensor_dim0_stride[31:0]
  g1[6] = 0;                                   // stride0[47:32]=0; stride1 lo=0
  g1[7] = 0;
  i32x4 z4 = {0, 0, 0, 0};
#if defined(__clang_major__) && (__clang_major__ >= 23)
  i32x8 z8 = {0, 0, 0, 0, 0, 0, 0, 0};
  __builtin_amdgcn_tensor_load_to_lds(g0, g1, z4, z4, z8, 0);
#else
  __builtin_amdgcn_tensor_load_to_lds(g0, g1, z4, z4, 0);
#endif
}

// Branch-free tanh: 1 - 2/(exp2(2*log2e*u) + 1). Saturates correctly at +-1
// via v_exp_f32 -> inf/0 and v_rcp_f32, no compares, no SALU control flow.
__device__ inline float tanh_fast(float u) {
  float e = __builtin_amdgcn_exp2f(u * 2.8853900817779268f);
  return 1.0f - 2.0f * __builtin_amdgcn_rcpf(e + 1.0f);
}

__global__ __launch_bounds__(TPB) void sde_diag_kernel(
    const float* __restrict__ z_start,
    const float* __restrict__ Wf1, const float* __restrict__ bf1,
    const float* __restrict__ Wf2, const float* __restrict__ bf2,
    const float* __restrict__ Wg1, const float* __restrict__ bg1,
    const float* __restrict__ Wg2, const float* __restrict__ bg2,
    const float* __restrict__ noise,
    float* __restrict__ out) {
  // Weight LDS layout (332 f32):
  //   [0,72)   Wf1 (3x24)   [72,96)  bf1
  //   [96,168) Wf2 (24x3)   [168,171) bf2
  //   [171,267) Wg1 (3x32)  [267,299) bg1
  //   [299,331) Wg2 (32)    [331] bg2
  __shared__ float wts[332];
  __shared__ float nbuf[2][TCHUNK * ROWW];

  const int tid  = threadIdx.x;
  const int lane = tid & (GRP - 1);              // lane within 8-lane group
  const int grp  = tid >> 3;                     // group index in block
  const int b    = blockIdx.x;
  const int s    = b * SPB + grp;                // this group's sample

  for (int i = tid; i < 72; i += TPB) wts[i]       = Wf1[i];
  for (int i = tid; i < 24; i += TPB) wts[72 + i]  = bf1[i];
  for (int i = tid; i < 72; i += TPB) wts[96 + i]  = Wf2[i];
  for (int i = tid; i < 3;  i += TPB) wts[168 + i] = bf2[i];
  for (int i = tid; i < 96; i += TPB) wts[171 + i] = Wg1[i];
  for (int i = tid; i < 32; i += TPB) wts[267 + i] = bg1[i];
  for (int i = tid; i < 32; i += TPB) wts[299 + i] = Wg2[i];
  if (tid == 0) wts[331] = bg2[0];

  // State replicated across the 8 lanes of each group.
  float y0 = z_start[s * 3 + 0];
  float y1 = z_start[s * 3 + 1];
  float y2 = z_start[s * 3 + 2];
  if (s == 0 && lane == 0) { out[0] = y0; out[1] = y1; out[2] = y2; }

  const float dt  = 1.0f / (float)(NSAMP - 1);
  const float sdt = sqrtf(dt);

  const int nsteps  = b * SPB + (SPB - 1);       // max step any group here needs
  const int nchunks = (nsteps + TCHUNK - 1) / TCHUNK;
  const u32 rowstride = NSAMP * DDIM;            // 12288 f32 between step rows

  // Prologue: wave 0 kicks off chunk 0.
  if (tid < 32) {
    int rows = nsteps < TCHUNK ? nsteps : TCHUNK;
    tdm_load_2d((u32)(uintptr_t)&nbuf[0][0],
                noise + (size_t)(b * SPB) * DDIM,
                ROWW, (u32)rows, rowstride, (u32)(NSAMP - 1), rowstride);
  }

  int t = 1;
  for (int c = 0; c < nchunks; ++c) {
    const int cur = c & 1;
    if (tid < 32) {
      if (c + 1 < nchunks) {
        // Issue chunk c+1 first (its buffer was last read in iteration c-1,
        // fenced by that iteration's trailing barrier), then wait for chunk c
        // only: TDM completes in order per wave, so TENSORcnt<=1 means chunk c
        // has landed while c+1's DMA overlaps this whole compute phase.
        const int done = (c + 1) * TCHUNK;
        int rows = nsteps - done; if (rows > TCHUNK) rows = TCHUNK;
        tdm_load_2d((u32)(uintptr_t)&nbuf[cur ^ 1][0],
                    noise + (size_t)done * rowstride + (size_t)(b * SPB) * DDIM,
                    ROWW, (u32)rows, rowstride, (u32)(NSAMP - 1), rowstride);
        __builtin_amdgcn_s_wait_tensorcnt(1);
      } else {
        __builtin_amdgcn_s_wait_tensorcnt(0);
      }
    }
    __syncthreads();                                     // publish chunk c

    int csteps = nsteps - c * TCHUNK; if (csteps > TCHUNK) csteps = TCHUNK;
    for (int r = 0; r < csteps; ++r, ++t) {
      if (t <= s) {            // uniform across the 8-lane group
        const float* dwp = &nbuf[cur][r * ROWW + grp * 3];
        const float dw0 = dwp[0], dw1 = dwp[1], dw2 = dwp[2];   // broadcast

        // drift partials: this lane handles hidden units lane, lane+8, lane+16
        float pf0 = 0.0f, pf1 = 0.0f, pf2 = 0.0f;
        #pragma unroll
        for (int k = 0; k < HF / GRP; ++k) {
          const int j = lane + GRP * k;
          float u = fmaf(y0, wts[j],
                    fmaf(y1, wts[24 + j],
                    fmaf(y2, wts[48 + j], wts[72 + j])));
          float h = tanh_fast(u);
          pf0 = fmaf(h, wts[96 + j * 3 + 0], pf0);
          pf1 = fmaf(h, wts[96 + j * 3 + 1], pf1);
          pf2 = fmaf(h, wts[96 + j * 3 + 2], pf2);
        }

        // diffusion partials: hidden units lane, lane+8, lane+16, lane+24
        float pacc = 0.0f;
        #pragma unroll
        for (int k = 0; k < HG / GRP; ++k) {
          const int j = lane + GRP * k;
          float u = fmaf(y0, wts[171 + j],
                    fmaf(y1, wts[203 + j],
                    fmaf(y2, wts[235 + j], wts[267 + j])));
          pacc = fmaf(fmaxf(u, 0.0f), wts[299 + j], pacc);
        }

        // butterfly reduce within the group; all 8 lanes end with full sums
        #pragma unroll
        for (int m = 1; m < GRP; m <<= 1) {
          pf0  += __shfl_xor(pf0,  m, 32);
          pf1  += __shfl_xor(pf1,  m, 32);
          pf2  += __shfl_xor(pf2,  m, 32);
          pacc += __shfl_xor(pacc, m, 32);
        }
        const float f0 = pf0 + wts[168];
        const float f1 = pf1 + wts[169];
        const float f2 = pf2 + wts[170];
        const float g  = 0.1f * __builtin_amdgcn_exp2f(
                             (pacc + wts[331]) * 1.4426950408889634f);

        y0 = fmaf(g, dw0 * sdt, fmaf(f0, dt, y0));
        y1 = fmaf(g, dw1 * sdt, fmaf(f1, dt, y1));
        y2 = fmaf(g, dw2 * sdt, fmaf(f2, dt, y2));

        if (t == s && lane == 0) {                       // diagonal element
          out[s * 3 + 0] = y0;
          out[s * 3 + 1] = y1;
          out[s * 3 + 2] = y2;
        }
      }
    }
    __syncthreads();   // everyone done reading nbuf[cur] before it is re-filled
  }
}

extern "C" void kernel_launch(void* const* d_in, const int* in_sizes, int n_in,
                              void* d_out, int out_size, void* d_ws, size_t ws_size,
                              hipStream_t stream) {
  const float* z_start = (const float*)d_in[0];
  const float* Wf1     = (const float*)d_in[1];
  const float* bf1     = (const float*)d_in[2];
  const float* Wf2     = (const float*)d_in[3];
  const float* bf2     = (const float*)d_in[4];
  const float* Wg1     = (const float*)d_in[5];
  const float* bg1     = (const float*)d_in[6];
  const float* Wg2     = (const float*)d_in[7];
  const float* bg2     = (const float*)d_in[8];
  const float* noise   = (const float*)d_in[9];
  float* out           = (float*)d_out;

  sde_diag_kernel<<<dim3(NSAMP / SPB), dim3(TPB), 0, stream>>>(
      z_start, Wf1, bf1, Wf2, bf2, Wg1, bg1, Wg2, bg2, noise, out);
}